// DdlgLayer_90443421319689
// MI455X (gfx1250) — compile-verified
//
#include <hip/hip_runtime.h>

// Problem constants (match reference): B=2048, IN=4096, OUT=4096, C=32
#define B_TOT    2048
#define IN_DIM   4096
#define OUT_DIM  4096
#define C_DIM    32

// Tiling: each block handles BTILE batch rows x OTILE outputs.
#define BTILE    8      // 8 x-rows in LDS = 8*16KB = 128KB (<= 320KB/WGP)
#define OTILE    512
#define NTHREADS 256

typedef int v4i __attribute__((ext_vector_type(4)));
typedef __attribute__((address_space(1))) v4i gv4i_t;   // global int4
typedef __attribute__((address_space(3))) v4i lv4i_t;   // LDS int4

// ---- CDNA5 async global->LDS copy (16B per lane), tracked by ASYNCcnt ----
__device__ __forceinline__ void async_copy_b128(const void* gsrc, void* ldst) {
#if __has_builtin(__builtin_amdgcn_global_load_async_to_lds_b128)
    __builtin_amdgcn_global_load_async_to_lds_b128(
        (gv4i_t*)gsrc, (lv4i_t*)ldst, /*offset=*/0, /*cpol=*/0);
#else
    unsigned lds_off = (unsigned)(unsigned long long)((lv4i_t*)ldst);
    asm volatile("global_load_async_to_lds_b128 %0, %1, off"
                 :: "v"(lds_off), "v"(gsrc) : "memory");
#endif
}

__device__ __forceinline__ void wait_async_zero() {
#if __has_builtin(__builtin_amdgcn_s_wait_asynccnt)
    __builtin_amdgcn_s_wait_asynccnt(0);
#else
    asm volatile("s_wait_asynccnt 0" ::: "memory");
#endif
}

__global__ __launch_bounds__(NTHREADS)
void ddlg_kernel(const float* __restrict__ x,
                 const float* __restrict__ w,
                 const int*   __restrict__ conn,
                 float*       __restrict__ out) {
    __shared__ float xs[BTILE * IN_DIM];   // 128 KB static LDS

    const int tid = threadIdx.x;
    const int o0  = blockIdx.x * OTILE;
    const int b0  = blockIdx.y * BTILE;

    // ---- Stage BTILE rows of x into LDS via async copy engine ----
    // BTILE*IN_DIM/4 = 8192 float4's; 256 lanes -> 32 issues per lane.
    constexpr int NV4     = BTILE * IN_DIM / 4;      // 8192
    constexpr int V4_ROW  = IN_DIM / 4;              // 1024 (power of two)
    #pragma unroll
    for (int i = 0; i < NV4 / NTHREADS; ++i) {
        const int k  = i * NTHREADS + tid;
        const int r  = k >> 10;          // k / V4_ROW
        const int c4 = k & (V4_ROW - 1); // k % V4_ROW
        const float* g = x + (size_t)(b0 + r) * IN_DIM + (size_t)c4 * 4;
        async_copy_b128((const void*)g, (void*)&xs[k * 4]);
    }
    wait_async_zero();
    __syncthreads();

    // ---- Compute: 2 outputs per thread, each reused across BTILE batches ----
    #pragma unroll
    for (int t = 0; t < OTILE / NTHREADS; ++t) {
        const int o = o0 + t * NTHREADS + tid;

        // Load this output's 32 connection indices into registers (L2-hot,
        // amortized over BTILE batch rows).
        int idx[C_DIM];
        const int4* ip = (const int4*)(conn + (size_t)o * C_DIM);
        #pragma unroll
        for (int k = 0; k < C_DIM / 4; ++k) {
            int4 v = ip[k];
            idx[4 * k + 0] = v.x;
            idx[4 * k + 1] = v.y;
            idx[4 * k + 2] = v.z;
            idx[4 * k + 3] = v.w;
        }

        // Inline softmax of the 4 per-output weights.
        float4 wv = *(const float4*)(w + (size_t)o * 4);
        float m  = fmaxf(fmaxf(wv.x, wv.y), fmaxf(wv.z, wv.w));
        float e0 = expf(wv.x - m);
        float e1 = expf(wv.y - m);
        float e2 = expf(wv.z - m);
        float e3 = expf(wv.w - m);
        float inv = 1.0f / (e0 + e1 + e2 + e3);
        e0 *= inv; e1 *= inv; e2 *= inv; e3 *= inv;

        #pragma unroll
        for (int r = 0; r < BTILE; ++r) {
            const float* row = &xs[r * IN_DIM];
            float f0 = row[idx[0]];
            float mn = f0;
            float mx = f0;
            float pr = f0;
            float cp = 1.0f - f0;
            #pragma unroll
            for (int j = 1; j < C_DIM; ++j) {
                float f = row[idx[j]];
                mn = fminf(mn, f);
                mx = fmaxf(mx, f);
                pr *= f;
                cp *= (1.0f - f);   // co-product accumulator
            }
            out[(size_t)(b0 + r) * OUT_DIM + o] =
                e0 * mn + e1 * mx + e2 * pr + e3 * (1.0f - cp);
        }
    }
}

extern "C" void kernel_launch(void* const* d_in, const int* in_sizes, int n_in,
                              void* d_out, int out_size, void* d_ws, size_t ws_size,
                              hipStream_t stream) {
    const float* x    = (const float*)d_in[0];   // (B, IN) fp32
    const float* w    = (const float*)d_in[1];   // (OUT, 4) fp32
    const int*   conn = (const int*)d_in[2];     // (OUT, C) int32
    float*       out  = (float*)d_out;           // (B, OUT) fp32

    dim3 grid(OUT_DIM / OTILE, B_TOT / BTILE);   // 8 x 256 = 2048 blocks
    dim3 block(NTHREADS);
    ddlg_kernel<<<grid, block, 0, stream>>>(x, w, conn, out);
}